// RoIAlign_4174708211916
// MI455X (gfx1250) — compile-verified
//
#include <hip/hip_runtime.h>
#include <hip/hip_bf16.h>

typedef __attribute__((ext_vector_type(2))) float v2f;
typedef __attribute__((ext_vector_type(8))) float v8f;

#define FN 2
#define FC 256
#define FH 200
#define FW 304
#define NROIS 2000
#define AH 7
#define AW 7
#define MDIM 49          // AH*AW
#define MP 64            // M padded to 4 tiles of 16
#define KDIM 196         // 14 rows * 14 cols sampled
#define OUT_PER_ROI (FC * MDIM)   // 12544

// ---------------------------------------------------------------------------
// Kernel 1: NCHW -> NHWC transpose (coalesced both sides via LDS tile)
// grid: (ceil(W/32)=10, C/32=8, N*H=400), block: 256 (32x8 logical)
// ---------------------------------------------------------------------------
__global__ void __launch_bounds__(256)
nchw_to_nhwc_kernel(const float* __restrict__ in, float* __restrict__ out) {
    __shared__ float tile[32][33];
    const int ny = blockIdx.z;
    const int n = ny / FH, y = ny % FH;
    const int x0 = blockIdx.x * 32;
    const int c0 = blockIdx.y * 32;
    const int tx = threadIdx.x & 31;
    const int ty = threadIdx.x >> 5;          // 0..7

    #pragma unroll
    for (int i = 0; i < 4; ++i) {
        const int c = c0 + ty + i * 8;
        const int x = x0 + tx;
        float v = 0.0f;
        if (x < FW)
            v = in[(((size_t)n * FC + c) * FH + y) * FW + x];  // coalesced in x
        tile[ty + i * 8][tx] = v;             // tile[c_local][x_local]
    }
    __syncthreads();
    #pragma unroll
    for (int i = 0; i < 4; ++i) {
        const int x = x0 + ty + i * 8;
        if (x < FW)
            out[(((size_t)n * FH + y) * FW + x) * FC + c0 + tx] = tile[tx][ty + i * 8]; // coalesced in c
    }
}

// ---------------------------------------------------------------------------
// Kernel 2: per-RoI GEMM  out[49x256] = W2[49x196] x G[196x256]
// via chained V_WMMA_F32_16X16X4_F32.  One block per RoI, 8 wave32.
// Patch gather uses GLOBAL_LOAD_ASYNC_TO_LDS_B128 (ASYNCcnt).
// Dynamic LDS: W2 (64*196 floats) + G (196*256 floats) = 250,880 B
// ---------------------------------------------------------------------------
__global__ void __launch_bounds__(256)
roialign_wmma_kernel(const float* __restrict__ nhwc, const float* __restrict__ rois,
                     const float* __restrict__ scale_p, float* __restrict__ out) {
    extern __shared__ float smem[];
    float* Wlds = smem;                 // [MP][KDIM]
    float* Glds = smem + MP * KDIM;     // [KDIM][FC]

    __shared__ int   s_yi[AH], s_xi[AW], s_b;
    __shared__ float s_wy[2 * AH], s_wx[2 * AW];

    const int r = blockIdx.x;
    const int t = threadIdx.x;

    // ---- geometry (thread 0) ----
    if (t == 0) {
        const float sc = *scale_p;
        const float* rr = rois + (size_t)r * 5;
        s_b = (int)rr[0];
        const float x1 = rr[1] * sc, y1 = rr[2] * sc;
        const float x2 = rr[3] * sc, y2 = rr[4] * sc;
        const float roi_w = fmaxf(x2 - x1 + 1.0f, 0.0f);
        const float roi_h = fmaxf(y2 - y1 + 1.0f, 0.0f);
        const float bh = roi_h / (float)(AH - 1);
        const float bw = roi_w / (float)(AW - 1);
        #pragma unroll
        for (int ph = 0; ph < AH; ++ph) {
            const float h  = y1 + (float)ph * bh;
            const bool  vh = (h >= 0.0f) && (h < (float)FH);
            const float hs = fminf(fmaxf(floorf(h), 0.0f), (float)(FH - 2));
            const float dh = h - hs;
            s_yi[ph]         = (int)hs;
            s_wy[2 * ph]     = vh ? (1.0f - dh) : 0.0f;
            s_wy[2 * ph + 1] = vh ? dh : 0.0f;
        }
        #pragma unroll
        for (int pw = 0; pw < AW; ++pw) {
            const float w  = x1 + (float)pw * bw;
            const bool  vw = (w >= 0.0f) && (w < (float)FW);
            const float ws = fminf(fmaxf(floorf(w), 0.0f), (float)(FW - 2));
            const float dw = w - ws;
            s_xi[pw]         = (int)ws;
            s_wx[2 * pw]     = vw ? (1.0f - dw) : 0.0f;
            s_wx[2 * pw + 1] = vw ? dw : 0.0f;
        }
    }
    __syncthreads();

    // ---- gather G: 196 rows x 256 channels, async global -> LDS (b128/lane),
    //      overlapped with the W2 weight build below ----
    {
        const int c4 = (t & 63) << 2;       // 16B per lane, 64 lanes per row
        const unsigned bbase = (unsigned)(s_b * FH);
        for (int k0 = 0; k0 < KDIM; k0 += 4) {
            const int k  = k0 + (t >> 6);   // 4 rows per iteration
            const int rk = k / 14, ck = k - rk * 14;
            const int y  = s_yi[rk >> 1] + (rk & 1);
            const int x  = s_xi[ck >> 1] + (ck & 1);
            const unsigned goff =
                (((bbase + (unsigned)y) * FW + (unsigned)x) * FC + (unsigned)c4) * 4u;
            const unsigned ldst =
                (unsigned)(uintptr_t)(Glds + (size_t)k * FC + c4);
            asm volatile("global_load_async_to_lds_b128 %0, %1, %2"
                         :: "v"(ldst), "v"(goff), "s"(nhwc)
                         : "memory");
        }
    }

    // ---- build W2 in LDS (padded rows m>=49 are zero) ----
    for (int idx = t; idx < MP * KDIM; idx += 256) {
        const int m = idx / KDIM;
        const int k = idx - m * KDIM;
        float v = 0.0f;
        if (m < MDIM) {
            const int ph = m / AW, pw = m - ph * AW;
            const int rk = k / 14, ck = k - rk * 14;
            if ((rk >> 1) == ph && (ck >> 1) == pw)
                v = s_wy[rk] * s_wx[ck];
        }
        Wlds[idx] = v;
    }

    asm volatile("s_wait_asynccnt 0x0" ::: "memory");
    __syncthreads();

    // ---- GEMM: 64 16x16 tiles (4 Mtiles x 16 Ntiles), 8 per wave ----
    const int wave = t >> 5;
    const int lane = t & 31;
    const int hi   = lane >> 4;     // selects K pair per A/B VGPR layout
    const int ln   = lane & 15;
    float* const out_r = out + (size_t)r * OUT_PER_ROI;

    for (int tile = wave; tile < 64; tile += 8) {   // uniform across all waves
        const int mt = tile >> 4;                   // 0..3
        const int nt = tile & 15;                   // 0..15
        const int m  = mt * 16 + ln;                // A row for this lane
        const int n  = nt * 16 + ln;                // B col for this lane
        v8f acc = {};
        for (int k0 = 0; k0 < KDIM; k0 += 4) {
            const int kA = k0 + 2 * hi;
            v2f a, b;
            a.x = Wlds[(size_t)m * KDIM + kA];
            a.y = Wlds[(size_t)m * KDIM + kA + 1];
            b.x = Glds[(size_t)kA * FC + n];
            b.y = Glds[(size_t)(kA + 1) * FC + n];
            acc = __builtin_amdgcn_wmma_f32_16x16x4_f32(
                false, a, false, b, (short)0, acc, false, false);
        }
        // D layout: lane holds (M = mt*16 + 8*hi + j, N = n) in acc[j]
        float* const orow = out_r + (size_t)n * MDIM;   // contiguous 49 floats
        if (mt < 3) {
            const int mbase = mt * 16 + 8 * hi;         // all 8 in-bounds
            #pragma unroll
            for (int j = 0; j < 8; ++j) orow[mbase + j] = acc[j];
        } else if (hi == 0) {
            orow[48] = acc[0];                          // only M=48 valid
        }
    }
}

// ---------------------------------------------------------------------------
// Fallback: direct NCHW gather (used only if workspace can't hold NHWC copy)
// ---------------------------------------------------------------------------
__global__ void __launch_bounds__(256)
roialign_direct_kernel(const float* __restrict__ feat, const float* __restrict__ rois,
                       const float* __restrict__ scale_p, float* __restrict__ out) {
    const size_t idx = (size_t)blockIdx.x * 256 + threadIdx.x;
    const size_t total = (size_t)NROIS * OUT_PER_ROI;
    if (idx >= total) return;
    const int r   = (int)(idx / OUT_PER_ROI);
    const int rem = (int)(idx % OUT_PER_ROI);
    const int c   = rem / MDIM;
    const int m   = rem - c * MDIM;
    const int ph  = m / AW, pw = m - ph * AW;

    const float sc = *scale_p;
    const float* rr = rois + (size_t)r * 5;
    const int b = (int)rr[0];
    const float x1 = rr[1] * sc, y1 = rr[2] * sc;
    const float x2 = rr[3] * sc, y2 = rr[4] * sc;
    const float bh = fmaxf(y2 - y1 + 1.0f, 0.0f) / (float)(AH - 1);
    const float bw = fmaxf(x2 - x1 + 1.0f, 0.0f) / (float)(AW - 1);
    const float h = y1 + (float)ph * bh;
    const float w = x1 + (float)pw * bw;
    const bool valid = (h >= 0.0f) && (h < (float)FH) && (w >= 0.0f) && (w < (float)FW);
    const float hs = fminf(fmaxf(floorf(h), 0.0f), (float)(FH - 2));
    const float ws = fminf(fmaxf(floorf(w), 0.0f), (float)(FW - 2));
    const float dh = h - hs, dw = w - ws;
    const int yi = (int)hs, xi = (int)ws;

    const float* p = feat + (((size_t)b * FC + c) * FH + yi) * FW + xi;
    const float v00 = p[0], v01 = p[1], v10 = p[FW], v11 = p[FW + 1];
    float val = v00 * (1.0f - dh) * (1.0f - dw) + v01 * (1.0f - dh) * dw +
                v10 * dh * (1.0f - dw) + v11 * dh * dw;
    out[idx] = valid ? val : 0.0f;
}

// ---------------------------------------------------------------------------
extern "C" void kernel_launch(void* const* d_in, const int* in_sizes, int n_in,
                              void* d_out, int out_size, void* d_ws, size_t ws_size,
                              hipStream_t stream) {
    const float* feat  = (const float*)d_in[0];
    const float* rois  = (const float*)d_in[1];
    const float* scale = (const float*)d_in[2];
    float* out = (float*)d_out;

    const size_t nhwc_bytes = (size_t)FN * FH * FW * FC * sizeof(float);  // 124.5 MB
    if (ws_size >= nhwc_bytes) {
        float* nhwc = (float*)d_ws;
        dim3 tgrid((FW + 31) / 32, FC / 32, FN * FH);
        nchw_to_nhwc_kernel<<<tgrid, 256, 0, stream>>>(feat, nhwc);

        const size_t lds_bytes = (size_t)(MP * KDIM + KDIM * FC) * sizeof(float); // 250,880 B
        roialign_wmma_kernel<<<NROIS, 256, lds_bytes, stream>>>(nhwc, rois, scale, out);
    } else {
        const size_t total = (size_t)NROIS * OUT_PER_ROI;
        roialign_direct_kernel<<<(unsigned)((total + 255) / 256), 256, 0, stream>>>(
            feat, rois, scale, out);
    }
}